// MultiBoxLoss_64484638982323
// MI455X (gfx1250) — compile-verified
//
#include <hip/hip_runtime.h>
#include <hip/hip_bf16.h>
#include <math.h>

#define IOU_THR 0.5f
#define NEG_POS_RATIO 3
#define MAXM 20          // matching loop cap (reference M = 20)
#define WPB 8            // waves (=anchors) per block in loss kernel

typedef float v2f __attribute__((ext_vector_type(2)));
typedef float v8f __attribute__((ext_vector_type(8)));

__device__ __forceinline__ float warp_sum_f(float v) {
    for (int o = 16; o; o >>= 1) v += __shfl_xor(v, o, 32);
    return v;
}
__device__ __forceinline__ int warp_sum_i(int v) {
    for (int o = 16; o; o >>= 1) v += __shfl_xor(v, o, 32);
    return v;
}
__device__ __forceinline__ float warp_max_f(float v) {
    for (int o = 16; o; o >>= 1) v = fmaxf(v, __shfl_xor(v, o, 32));
    return v;
}

// ---------------------------------------------------------------------------
// K1: per-batch anchor<->GT matching.
// grid = B blocks, 256 threads. Each thread strides anchors; 20 GT boxes in LDS.
// Outputs: a_best_iou/idx (B,A), gt_best_iou/idx (B,M).
// ---------------------------------------------------------------------------
__global__ void match_kernel(const float* __restrict__ gt_boxes,   // (B,M,4) xyxy
                             const float* __restrict__ anchors,    // (A,4) cxcywh
                             float* __restrict__ a_best_iou,
                             int*   __restrict__ a_best_idx,
                             float* __restrict__ gt_best_iou,
                             int*   __restrict__ gt_best_idx,
                             int A, int M) {
    const int b = blockIdx.x;
    const int tid = threadIdx.x, lane = tid & 31, w = tid >> 5;
    __shared__ float4 sgt[MAXM];
    __shared__ float  sarea[MAXM];
    __shared__ float  riou[8][MAXM];
    __shared__ int    ridx[8][MAXM];

    const int MM = (M < MAXM) ? M : MAXM;
    if (tid < MM) {
        const float* g = gt_boxes + ((size_t)b * M + tid) * 4;
        float4 bb = make_float4(g[0], g[1], g[2], g[3]);
        sgt[tid] = bb;
        sarea[tid] = (bb.z - bb.x) * (bb.w - bb.y);
    }
    __syncthreads();

    float giou[MAXM];
    int   gidx[MAXM];
#pragma unroll
    for (int j = 0; j < MAXM; j++) { giou[j] = -1.0f; gidx[j] = 0x7fffffff; }

    for (int a = tid; a < A; a += blockDim.x) {
        const float* ap = anchors + (size_t)a * 4;
        float acx = ap[0], acy = ap[1], aw = ap[2], ah = ap[3];
        float ax1 = acx - 0.5f * aw, ay1 = acy - 0.5f * ah;
        float ax2 = acx + 0.5f * aw, ay2 = acy + 0.5f * ah;
        float aarea = aw * ah;
        float best = -1.0f; int bidx = 0;
#pragma unroll
        for (int j = 0; j < MAXM; j++) {
            if (j < MM) {
                float4 g = sgt[j];
                float ix = fmaxf(fminf(ax2, g.z) - fmaxf(ax1, g.x), 0.0f);
                float iy = fmaxf(fminf(ay2, g.w) - fmaxf(ay1, g.y), 0.0f);
                float inter = ix * iy;
                float iou = inter / (sarea[j] + aarea - inter + 1e-7f);
                if (iou > best) { best = iou; bidx = j; }   // first-index on ties
                if (iou > giou[j] || (iou == giou[j] && a < gidx[j])) {
                    giou[j] = iou; gidx[j] = a;
                }
            }
        }
        a_best_iou[(size_t)b * A + a] = best;
        a_best_idx[(size_t)b * A + a] = bidx;
    }

    // reduce per-GT (max IoU, smallest anchor idx on tie) across the block
#pragma unroll
    for (int j = 0; j < MAXM; j++) {
        if (j < MM) {
            float v = giou[j]; int id = gidx[j];
            for (int o = 16; o; o >>= 1) {
                float ov = __shfl_down(v, o, 32);
                int   oi = __shfl_down(id, o, 32);
                if (ov > v || (ov == v && oi < id)) { v = ov; id = oi; }
            }
            if (lane == 0) { riou[w][j] = v; ridx[w][j] = id; }
        }
    }
    __syncthreads();
    const int nw = blockDim.x >> 5;
    if (tid < MM) {
        float v = -2.0f; int id = 0x7fffffff;
        for (int i = 0; i < nw; i++) {
            float ov = riou[i][tid]; int oi = ridx[i][tid];
            if (ov > v || (ov == v && oi < id)) { v = ov; id = oi; }
        }
        gt_best_iou[(size_t)b * M + tid] = v;
        gt_best_idx[(size_t)b * M + tid] = id;
    }
}

// ---------------------------------------------------------------------------
// K2: sequential force-assignment (faithful to the reference fori_loop).
// One thread per batch. Pre-reads original a_best_iou before any writes.
// Marks forced positives by setting a_best_iou = 2.0 (> IOU_THR sentinel).
// ---------------------------------------------------------------------------
__global__ void force_assign_kernel(const float* __restrict__ gt_best_iou,
                                    const int*   __restrict__ gt_best_idx,
                                    float* __restrict__ a_best_iou,
                                    int*   __restrict__ a_best_idx,
                                    int A, int M) {
    if (threadIdx.x != 0) return;
    const int b = blockIdx.x;
    const int MM = (M < MAXM) ? M : MAXM;
    float orig[MAXM], gi[MAXM];
    int   ba[MAXM];
    for (int j = 0; j < MM; j++) {
        ba[j]   = gt_best_idx[(size_t)b * M + j];
        gi[j]   = gt_best_iou[(size_t)b * M + j];
        orig[j] = a_best_iou[(size_t)b * A + ba[j]];   // original (pre-modification)
    }
    for (int j = 0; j < MM; j++) {
        if (gi[j] > 1e-5f) {
            a_best_iou[(size_t)b * A + ba[j]] = 2.0f;              // force positive
            if (orig[j] < gi[j]) a_best_idx[(size_t)b * A + ba[j]] = j;  // sequential override
        }
    }
}

// ---------------------------------------------------------------------------
// K3: per-anchor cross-entropy (wave-per-anchor, coalesced 81-class rows),
// SmoothL1 for positives, per-block deterministic partials (no float atomics).
// Stores conf key (float bits, 0 for positives) for hard-negative mining.
// ---------------------------------------------------------------------------
__global__ void loss_kernel(const float* __restrict__ plocs,     // (B,A,4)
                            const float* __restrict__ scores,    // (B,A,C)
                            const float* __restrict__ gt_boxes,  // (B,M,4)
                            const int*   __restrict__ gt_labels, // (B,M)
                            const float* __restrict__ anchors,   // (A,4)
                            const float* __restrict__ a_best_iou,
                            const int*   __restrict__ a_best_idx,
                            unsigned* __restrict__ conf_key,
                            float* __restrict__ blk_loc,
                            float* __restrict__ blk_conf,
                            int*   __restrict__ blk_np,
                            int A, int M, int C) {
    const int b = blockIdx.y;
    const int w = threadIdx.x >> 5, lane = threadIdx.x & 31;
    const int a = blockIdx.x * WPB + w;
    __shared__ float s_loc[WPB], s_conf[WPB];
    __shared__ int   s_np[WPB];

    float my_loc = 0.0f, my_conf = 0.0f;
    int   my_np = 0;

    if (a < A) {
        const size_t idx = (size_t)b * A + a;
        const float* row = scores + idx * C;
        if (a + WPB < A) __builtin_prefetch(row + (size_t)WPB * C, 0, 1); // global_prefetch_b8
        const float NEG_INF = -__builtin_inff();
        float v0 = (lane < C)      ? row[lane]      : NEG_INF;
        float v1 = (lane + 32 < C) ? row[lane + 32] : NEG_INF;
        float v2 = (lane + 64 < C) ? row[lane + 64] : NEG_INF;
        float m = warp_max_f(fmaxf(fmaxf(v0, v1), v2));
        float s = 0.0f;
        if (lane < C)      s += expf(v0 - m);
        if (lane + 32 < C) s += expf(v1 - m);
        if (lane + 64 < C) s += expf(v2 - m);
        s = warp_sum_f(s);

        if (lane == 0) {
            bool pos = a_best_iou[idx] > IOU_THR;
            int  j   = a_best_idx[idx];
            int  tc  = pos ? (gt_labels[(size_t)b * M + j] + 1) : 0;
            float conf = (m + logf(s)) - row[tc];
            if (pos) {
                my_conf = conf; my_np = 1;
                conf_key[idx] = 0u;                    // never selected as negative
                // SmoothL1 vs encoded matched GT
                const float* g  = gt_boxes + ((size_t)b * M + j) * 4;
                float x1 = g[0], y1 = g[1], x2 = g[2], y2 = g[3];
                float mcx = 0.5f * (x1 + x2), mcy = 0.5f * (y1 + y2);
                float mw = x2 - x1, mh = y2 - y1;
                const float* ap = anchors + (size_t)a * 4;
                float e0 = (mcx - ap[0]) / ap[2];
                float e1 = (mcy - ap[1]) / ap[3];
                float e2 = logf(mw / ap[2] + 1e-7f);
                float e3 = logf(mh / ap[3] + 1e-7f);
                const float* p = plocs + idx * 4;
                float e[4] = {e0, e1, e2, e3};
                float ll = 0.0f;
                for (int c2 = 0; c2 < 4; c2++) {
                    float d = fabsf(p[c2] - e[c2]);
                    ll += (d < 1.0f) ? 0.5f * d * d : d - 0.5f;
                }
                my_loc = ll;
            } else {
                conf_key[idx] = __float_as_uint(fmaxf(conf, 0.0f)); // monotone uint key
            }
        }
    }
    if (lane == 0) { s_loc[w] = my_loc; s_conf[w] = my_conf; s_np[w] = my_np; }
    __syncthreads();
    if (threadIdx.x == 0) {
        float tl = 0.0f, tc2 = 0.0f; int tn = 0;
        for (int i = 0; i < WPB; i++) { tl += s_loc[i]; tc2 += s_conf[i]; tn += s_np[i]; }
        size_t bo = (size_t)b * gridDim.x + blockIdx.x;
        blk_loc[bo] = tl; blk_conf[bo] = tc2; blk_np[bo] = tn;   // deterministic writes
    }
}

// ---------------------------------------------------------------------------
// K4: per-batch reduce of partials + hard-negative mining via exact 32-step
// binary search on float bit keys (L2-resident per-batch key array).
// ---------------------------------------------------------------------------
__global__ void mine_kernel(const unsigned* __restrict__ conf_key,
                            const float* __restrict__ blk_loc,
                            const float* __restrict__ blk_conf,
                            const int*   __restrict__ blk_np,
                            float* __restrict__ loc_out,
                            float* __restrict__ conf_out,
                            int*   __restrict__ np_out,
                            int A, int M, int NBX) {
    const int b = blockIdx.x;
    const int tid = threadIdx.x, lane = tid & 31, w = tid >> 5;
    const int nw = blockDim.x >> 5;
    __shared__ float rf[32], rf2[32];
    __shared__ int   ri[32];
    __shared__ int   s_cnt, s_np;
    __shared__ float s_locb, s_confp;

    // stage 1: deterministic reduce of per-block partials
    float tl = 0.0f, tcf = 0.0f; int tn = 0;
    for (int i = tid; i < NBX; i += blockDim.x) {
        tl  += blk_loc [(size_t)b * NBX + i];
        tcf += blk_conf[(size_t)b * NBX + i];
        tn  += blk_np  [(size_t)b * NBX + i];
    }
    tl = warp_sum_f(tl); tcf = warp_sum_f(tcf); tn = warp_sum_i(tn);
    if (lane == 0) { rf[w] = tl; rf2[w] = tcf; ri[w] = tn; }
    __syncthreads();
    if (tid == 0) {
        float a0 = 0.0f, a1 = 0.0f; int a2 = 0;
        for (int i = 0; i < nw; i++) { a0 += rf[i]; a1 += rf2[i]; a2 += ri[i]; }
        s_locb = a0; s_confp = a1; s_np = a2;
    }
    __syncthreads();
    const int np = s_np;
    const int k = (np > 0) ? min(NEG_POS_RATIO * np, A - np)
                           : min(NEG_POS_RATIO * M, A);
    const unsigned* key = conf_key + (size_t)b * A;

    float neg_sum = 0.0f;
    if (k > 0) {
        unsigned lo = 0u, hi = 0xFFFFFFFFu;
        while (lo < hi) {                       // <= 32 iterations, uniform
            unsigned mid = lo + ((hi - lo) >> 1);
            int c = 0;
            for (int a = tid; a < A; a += blockDim.x) c += (key[a] > mid) ? 1 : 0;
            c = warp_sum_i(c);
            if (lane == 0) ri[w] = c;
            __syncthreads();
            if (tid == 0) { int t = 0; for (int i = 0; i < nw; i++) t += ri[i]; s_cnt = t; }
            __syncthreads();
            if (s_cnt >= k) lo = mid + 1u; else hi = mid;
            __syncthreads();
        }
        const unsigned v = lo;                  // k-th largest key (exact)
        int c = 0; float sgt = 0.0f;
        for (int a = tid; a < A; a += blockDim.x) {
            unsigned kk = key[a];
            if (kk > v) { c++; sgt += __uint_as_float(kk); }
        }
        c = warp_sum_i(c); sgt = warp_sum_f(sgt);
        if (lane == 0) { ri[w] = c; rf[w] = sgt; }
        __syncthreads();
        if (tid == 0) {
            int tc2 = 0; float ts = 0.0f;
            for (int i = 0; i < nw; i++) { tc2 += ri[i]; ts += rf[i]; }
            neg_sum = ts + (float)(k - tc2) * __uint_as_float(v); // ties at v
        }
    }
    if (tid == 0) {
        loc_out[b]  = s_locb;
        conf_out[b] = s_confp + neg_sum;
        np_out[b]   = np;
    }
}

// ---------------------------------------------------------------------------
// K5: final cross-batch reduction using V_WMMA_F32_16X16X4_F32 (f32 exact).
// B-matrix packs 32 loc partials in K rows 0..1 and 32 conf partials in rows
// 2..3; selector A rows [1,1,0,0] / [0,0,1,1] yield both column-sums in D
// rows 0 and 1, finished with a wave-shuffle sum. One wave, EXEC all ones.
// ---------------------------------------------------------------------------
__global__ void final_kernel(const float* __restrict__ loc_in,
                             const float* __restrict__ conf_in,
                             const int*   __restrict__ np_in,
                             float* __restrict__ out, int B) {
    const int lane = threadIdx.x;   // blockDim == 32
    float loc = 0.0f, conf = 0.0f, np = 0.0f;
    for (int b = lane; b < B; b += 32) {
        loc  += loc_in[b];
        conf += conf_in[b];
        np   += (float)np_in[b];
    }
    // B-matrix (4x16 f32, 2 VGPRs): b0 = {B[0][n] | B[2][n]}, b1 = {B[1][n] | B[3][n]}
    float conf_lm16 = __shfl(conf, (lane - 16) & 31, 32);
    float loc_lp16  = __shfl(loc,  (lane + 16) & 31, 32);
    v2f Bm; Bm.x = (lane < 16) ? loc : conf_lm16;
            Bm.y = (lane < 16) ? loc_lp16 : conf;
    // A-matrix (16x4 selector): row0=[1,1,0,0], row1=[0,0,1,1]
    float sel = (lane == 0 || lane == 17) ? 1.0f : 0.0f;
    v2f Am; Am.x = sel; Am.y = sel;
    v8f c = {0.0f, 0.0f, 0.0f, 0.0f, 0.0f, 0.0f, 0.0f, 0.0f};
    c = __builtin_amdgcn_wmma_f32_16x16x4_f32(
            /*neg_a=*/false, Am, /*neg_b=*/false, Bm,
            /*c_mod=*/(short)0, c, /*reuse_a=*/false, /*reuse_b=*/false);
    // D rows 2..15 are zero => full-wave shuffle sums are exact totals.
    float loc_total  = warp_sum_f(c[0]);   // D[0][n] = loc[n] + loc[n+16]
    float conf_total = warp_sum_f(c[1]);   // D[1][n] = conf[n] + conf[n+16]
    float np_total   = warp_sum_f(np);
    if (lane == 0) {
        float nf = fmaxf(np_total, 1.0f);
        out[0] = (loc_total + conf_total) / nf;
        out[1] = loc_total / nf;
        out[2] = conf_total / nf;
    }
}

// ---------------------------------------------------------------------------
extern "C" void kernel_launch(void* const* d_in, const int* in_sizes, int n_in,
                              void* d_out, int out_size, void* d_ws, size_t ws_size,
                              hipStream_t stream) {
    const float* plocs     = (const float*)d_in[0];   // (B,A,4)
    const float* scores    = (const float*)d_in[1];   // (B,A,C)
    const float* gt_boxes  = (const float*)d_in[2];   // (B,M,4)
    const int*   gt_labels = (const int*)  d_in[3];   // (B,M)
    const float* anchors   = (const float*)d_in[4];   // (A,4)

    const int A  = in_sizes[4] / 4;
    const int Bn = in_sizes[0] / (A * 4);
    const int C  = (int)((long long)in_sizes[1] / ((long long)Bn * A));
    const int M  = in_sizes[3] / Bn;
    const int NBX = (A + WPB - 1) / WPB;

    char* p = (char*)d_ws;
    float*    a_best_iou = (float*)p;    p += (size_t)Bn * A * sizeof(float);
    int*      a_best_idx = (int*)p;      p += (size_t)Bn * A * sizeof(int);
    unsigned* conf_key   = (unsigned*)p; p += (size_t)Bn * A * sizeof(unsigned);
    float*    gt_biou    = (float*)p;    p += (size_t)Bn * M * sizeof(float);
    int*      gt_bidx    = (int*)p;      p += (size_t)Bn * M * sizeof(int);
    float*    blk_loc    = (float*)p;    p += (size_t)Bn * NBX * sizeof(float);
    float*    blk_conf   = (float*)p;    p += (size_t)Bn * NBX * sizeof(float);
    int*      blk_np     = (int*)p;      p += (size_t)Bn * NBX * sizeof(int);
    float*    loc_b      = (float*)p;    p += (size_t)Bn * sizeof(float);
    float*    conf_b     = (float*)p;    p += (size_t)Bn * sizeof(float);
    int*      np_b       = (int*)p;      p += (size_t)Bn * sizeof(int);
    (void)ws_size; (void)n_in; (void)out_size;

    match_kernel<<<Bn, 256, 0, stream>>>(gt_boxes, anchors, a_best_iou, a_best_idx,
                                         gt_biou, gt_bidx, A, M);
    force_assign_kernel<<<Bn, 32, 0, stream>>>(gt_biou, gt_bidx, a_best_iou, a_best_idx, A, M);
    dim3 g3(NBX, Bn);
    loss_kernel<<<g3, 32 * WPB, 0, stream>>>(plocs, scores, gt_boxes, gt_labels, anchors,
                                             a_best_iou, a_best_idx, conf_key,
                                             blk_loc, blk_conf, blk_np, A, M, C);
    mine_kernel<<<Bn, 1024, 0, stream>>>(conf_key, blk_loc, blk_conf, blk_np,
                                         loc_b, conf_b, np_b, A, M, NBX);
    final_kernel<<<1, 32, 0, stream>>>(loc_b, conf_b, np_b, (float*)d_out, Bn);
}